// Classifier_45861660787100
// MI455X (gfx1250) — compile-verified
//
#include <hip/hip_runtime.h>
#include <hip/hip_bf16.h>

#define N_NODES   100000
#define N_EDGES   1600000
#define HID       128
#define N_GRAPHS  64
#define N_CLASSES 10

typedef __attribute__((ext_vector_type(2))) float v2f;
typedef __attribute__((ext_vector_type(8))) float v8f;

// ---------------------------------------------------------------- zero fill
__global__ void zero_f32(float* __restrict__ p, int n) {
    int i = blockIdx.x * blockDim.x + threadIdx.x;
    if (i < n) p[i] = 0.0f;
}

// ---------------------------------------------------------------- degrees
__global__ void compute_deg(const int* __restrict__ src, const int* __restrict__ dst,
                            float* __restrict__ deg_out, float* __restrict__ deg_in,
                            int nedges) {
    int e = blockIdx.x * blockDim.x + threadIdx.x;
    if (e < nedges) {
        atomicAdd(&deg_out[src[e]], 1.0f);
        atomicAdd(&deg_in[dst[e]], 1.0f);
    }
}

// deg -> (deg>0 ? rsqrt(deg) : 0), in place over a contiguous span
__global__ void finalize_rsqrt(float* __restrict__ deg, int n) {
    int i = blockIdx.x * blockDim.x + threadIdx.x;
    if (i < n) {
        float d = deg[i];
        deg[i] = (d > 0.0f) ? rsqrtf(d) : 0.0f;
    }
}

// ---------------------------------------------------------------- edge scatter
// agg[dst] += feat[src] * dinv_out[src]; one wave32 per edge, lane owns 4 dims.
__global__ __launch_bounds__(256) void scatter_edges(
        const int* __restrict__ src, const int* __restrict__ dst,
        const float* __restrict__ feat, const float* __restrict__ dinv_out,
        float* __restrict__ agg, int nedges) {
    int wave = threadIdx.x >> 5;
    int lane = threadIdx.x & 31;
    int e = blockIdx.x * 8 + wave;
    if (e >= nedges) return;
    int s = src[e];
    int d = dst[e];
    float sc = dinv_out[s];
    float4 v = ((const float4*)(feat + (size_t)s * HID))[lane];
    float* dp = agg + (size_t)d * HID + lane * 4;
    atomicAdd(dp + 0, v.x * sc);
    atomicAdd(dp + 1, v.y * sc);
    atomicAdd(dp + 2, v.z * sc);
    atomicAdd(dp + 3, v.w * sc);
}

// ---------------------------------------------------------------- fused GEMM
// out = relu( (diag(dinv) * A) @ W + bias ), A:[nrows x 128], W:[128 x 128].
// 256 threads = 8 waves; wave w computes rows [blk*128 + w*16, +16) x all 128 cols
// with V_WMMA_F32_16X16X4_F32. W staged through LDS in 32-row K-chunks.
__global__ __launch_bounds__(256) void gemm_bias_relu(
        const float* __restrict__ A, const float* __restrict__ dinv,
        const float* __restrict__ W, const float* __restrict__ bias,
        float* __restrict__ out, int nrows) {
    __shared__ float lw[32][132];            // K-chunk x N, padded vs bank conflicts

    int tid  = threadIdx.x;
    int wave = tid >> 5;
    int lane = tid & 31;
    int m    = lane & 15;                    // row-in-tile / col-in-tile index
    int half = lane >> 4;                    // lane half selects K pair (ISA layout)

    int rowA  = blockIdx.x * 128 + wave * 16 + m;     // A-fragment row this lane feeds
    float scale = (rowA < nrows) ? dinv[rowA] : 0.0f; // OOB rows contribute 0
    int rowAc = rowA < nrows ? rowA : (nrows - 1);    // clamped, always-valid address
    const float* arow = A + (size_t)rowAc * HID;

    v8f acc[8] = {};                                  // 16 x 128 output slab

    for (int kc = 0; kc < HID; kc += 32) {
        __syncthreads();
        // stage W[kc..kc+31][0..127] : 1024 float4 loads across 256 threads
        for (int i = tid; i < 32 * 32; i += 256) {
            int kk  = i >> 5;
            int nn4 = i & 31;
            float4 v = ((const float4*)(W + (size_t)(kc + kk) * HID))[nn4];
            lw[kk][nn4 * 4 + 0] = v.x;
            lw[kk][nn4 * 4 + 1] = v.y;
            lw[kk][nn4 * 4 + 2] = v.z;
            lw[kk][nn4 * 4 + 3] = v.w;
        }
        __syncthreads();

        #pragma unroll
        for (int k0 = 0; k0 < 32; k0 += 4) {
            int kb = k0 + 2 * half;          // lanes 0-15: K={k0,k0+1}; 16-31: {k0+2,k0+3}
            float2 av = *(const float2*)(arow + kc + kb);
            v2f a;
            a.x = av.x * scale;
            a.y = av.y * scale;
            #pragma unroll
            for (int t = 0; t < 8; ++t) {    // 8 column tiles of 16
                v2f b;
                b.x = lw[kb + 0][t * 16 + m];
                b.y = lw[kb + 1][t * 16 + m];
                acc[t] = __builtin_amdgcn_wmma_f32_16x16x4_f32(
                    false, a, false, b, (short)0, acc[t], false, false);
            }
        }
    }

    // epilogue: D vgpr j -> row (j + 8*half) of the tile, col = t*16 + m
    #pragma unroll
    for (int t = 0; t < 8; ++t) {
        int col = t * 16 + m;
        float bv = bias[col];
        #pragma unroll
        for (int j = 0; j < 8; ++j) {
            int r = blockIdx.x * 128 + wave * 16 + j + 8 * half;
            if (r < nrows) {
                float v = acc[t][j] + bv;
                out[(size_t)r * HID + col] = v > 0.0f ? v : 0.0f;
            }
        }
    }
}

// ---------------------------------------------------------------- mean pool (sum part)
__global__ __launch_bounds__(256) void pool_sum(
        const float* __restrict__ h, const int* __restrict__ gid,
        float* __restrict__ sums, float* __restrict__ cnt, int nnodes) {
    int wave = threadIdx.x >> 5;
    int lane = threadIdx.x & 31;
    int node = blockIdx.x * 8 + wave;
    if (node >= nnodes) return;
    int g = gid[node];
    float4 v = ((const float4*)(h + (size_t)node * HID))[lane];
    float* sp = sums + (size_t)g * HID + lane * 4;
    atomicAdd(sp + 0, v.x);
    atomicAdd(sp + 1, v.y);
    atomicAdd(sp + 2, v.z);
    atomicAdd(sp + 3, v.w);
    if (lane == 0) atomicAdd(&cnt[g], 1.0f);
}

// ---------------------------------------------------------------- classifier head
__global__ void classify(const float* __restrict__ sums, const float* __restrict__ cnt,
                         const float* __restrict__ Wc, const float* __restrict__ bc,
                         float* __restrict__ out) {
    int t = threadIdx.x;
    if (t >= N_GRAPHS * N_CLASSES) return;
    int g = t / N_CLASSES;
    int c = t % N_CLASSES;
    float inv = 1.0f / fmaxf(cnt[g], 1.0f);
    float acc = bc[c];
    for (int k = 0; k < HID; ++k)
        acc = fmaf(sums[g * HID + k] * inv, Wc[k * N_CLASSES + c], acc);
    out[g * N_CLASSES + c] = acc;
}

// ---------------------------------------------------------------- launcher
extern "C" void kernel_launch(void* const* d_in, const int* in_sizes, int n_in,
                              void* d_out, int out_size, void* d_ws, size_t ws_size,
                              hipStream_t stream) {
    const float* h   = (const float*)d_in[0];
    const int*   src = (const int*)d_in[1];
    const int*   dst = (const int*)d_in[2];
    const int*   gid = (const int*)d_in[3];
    const float* W1  = (const float*)d_in[4];
    const float* b1  = (const float*)d_in[5];
    const float* W2  = (const float*)d_in[6];
    const float* b2  = (const float*)d_in[7];
    const float* Wc  = (const float*)d_in[8];
    const float* bc  = (const float*)d_in[9];
    float* out = (float*)d_out;

    const size_t FEAT_BYTES = (size_t)N_NODES * HID * sizeof(float);   // 51,200,000
    char* ws = (char*)d_ws;
    float* agg      = (float*)(ws);                                    // [N x 128]
    float* hbuf     = (float*)(ws + FEAT_BYTES);                       // [N x 128]
    float* dout_inv = (float*)(ws + 2 * FEAT_BYTES);                   // [N]
    float* din_inv  = (float*)(ws + 2 * FEAT_BYTES + (size_t)N_NODES * 4);
    float* psum     = (float*)(ws + 2 * FEAT_BYTES + 2 * (size_t)N_NODES * 4);
    float* pcnt     = psum + N_GRAPHS * HID;

    const int featN   = N_NODES * HID;                      // 12.8M floats
    const int smallN  = 2 * N_NODES + N_GRAPHS * HID + N_GRAPHS; // degs+pool span
    const int zb      = 256;

    // 1) zero degree + pool scratch (contiguous), then degrees -> rsqrt
    zero_f32<<<(smallN + zb - 1) / zb, zb, 0, stream>>>(dout_inv, smallN);
    compute_deg<<<(N_EDGES + zb - 1) / zb, zb, 0, stream>>>(src, dst, dout_inv, din_inv, N_EDGES);
    finalize_rsqrt<<<(2 * N_NODES + zb - 1) / zb, zb, 0, stream>>>(dout_inv, 2 * N_NODES);

    // 2) layer 1: scatter then fused normalize+GEMM+bias+relu
    zero_f32<<<(featN + zb - 1) / zb, zb, 0, stream>>>(agg, featN);
    scatter_edges<<<N_EDGES / 8, 256, 0, stream>>>(src, dst, h, dout_inv, agg, N_EDGES);
    gemm_bias_relu<<<(N_NODES + 127) / 128, 256, 0, stream>>>(agg, din_inv, W1, b1, hbuf, N_NODES);

    // 3) layer 2 (hbuf read by scatter, then overwritten in place by GEMM output)
    zero_f32<<<(featN + zb - 1) / zb, zb, 0, stream>>>(agg, featN);
    scatter_edges<<<N_EDGES / 8, 256, 0, stream>>>(src, dst, hbuf, dout_inv, agg, N_EDGES);
    gemm_bias_relu<<<(N_NODES + 127) / 128, 256, 0, stream>>>(agg, din_inv, W2, b2, hbuf, N_NODES);

    // 4) mean pool + classifier
    pool_sum<<<(N_NODES + 7) / 8, 256, 0, stream>>>(hbuf, gid, psum, pcnt, N_NODES);
    classify<<<1, N_GRAPHS * N_CLASSES, 0, stream>>>(psum, pcnt, Wc, bc, out);
}